// GraphConv_18691697672406
// MI455X (gfx1250) — compile-verified
//
#include <hip/hip_runtime.h>

typedef __attribute__((ext_vector_type(2))) float v2f;
typedef __attribute__((ext_vector_type(8))) float v8f;

static constexpr int DF = 128;   // feature dim (D_IN == D_OUT == 128)

// ---------------------------------------------------------------------------
// Kernel 1: support = x @ W + lin_bias   via V_WMMA_F32_16X16X4_F32
// One wave computes a 16-row strip across all 128 output columns (8 N-tiles).
// W is staged transposed in LDS so B fragments are contiguous 8B ds reads.
// ---------------------------------------------------------------------------
__global__ __launch_bounds__(256) void gemm_bias_wmma(
    const float* __restrict__ x, const float* __restrict__ W,
    const float* __restrict__ lin_bias, float* __restrict__ support,
    int nNodes, int mTiles)
{
    __shared__ float wt[DF * DF];           // wt[n*128 + k] = W[k*128 + n]

    for (int idx = threadIdx.x; idx < DF * DF; idx += blockDim.x) {
        int k = idx >> 7;
        int n = idx & 127;
        wt[n * DF + k] = W[idx];
    }
    __syncthreads();

    const int waveInBlock = threadIdx.x >> 5;
    const int lane        = threadIdx.x & 31;
    const int mtile       = blockIdx.x * (blockDim.x >> 5) + waveInBlock;
    if (mtile >= mTiles) return;            // wave-uniform: EXEC stays all-1s

    const int m0    = mtile << 4;
    const int mrow  = lane & 15;            // row within A tile / col within B tile
    const int khalf = lane >> 4;            // 0 or 1: which K pair this lane holds

    int mglob = m0 + mrow;
    if (mglob >= nNodes) mglob = nNodes - 1;        // clamp (tail safety)
    const float* __restrict__ xrow = x + (size_t)mglob * DF;

    v8f acc[8] = {};                        // 8 N-tiles of 16x16 f32 accum

    for (int k0 = 0; k0 < DF; k0 += 4) {
        const int kk = k0 + 2 * khalf;
        v2f a;                              // A 16x4: VGPR j, lane l -> A[l%16][k0+2*(l/16)+j]
        a.x = xrow[kk];
        a.y = xrow[kk + 1];
        #pragma unroll
        for (int nt = 0; nt < 8; ++nt) {
            const int n = (nt << 4) + mrow;
            v2f b;                          // B 4x16: VGPR j, lane l -> B[k0+2*(l/16)+j][l%16]
            b.x = wt[n * DF + kk];
            b.y = wt[n * DF + kk + 1];
            acc[nt] = __builtin_amdgcn_wmma_f32_16x16x4_f32(
                false, a, false, b, (short)0, acc[nt], false, false);
        }
    }

    // C/D layout: VGPR j, lane l -> C[j + 8*(l/16)][l%16]
    #pragma unroll
    for (int nt = 0; nt < 8; ++nt) {
        const int n  = (nt << 4) + mrow;
        const float lb = lin_bias[n];
        #pragma unroll
        for (int j = 0; j < 8; ++j) {
            const int m = m0 + j + 8 * khalf;
            if (m < nNodes)
                support[(size_t)m * DF + n] = acc[nt][j] + lb;
        }
    }
}

// ---------------------------------------------------------------------------
// CSR build phase: histogram -> exclusive scan -> permutation fill.
// ---------------------------------------------------------------------------
__global__ __launch_bounds__(256) void zero_ints(int* __restrict__ p, int n)
{
    int i = blockIdx.x * blockDim.x + threadIdx.x;
    if (i < n) p[i] = 0;
}

__global__ __launch_bounds__(256) void hist_rows(
    const int* __restrict__ rows, int* __restrict__ rowCnt, int nEdges)
{
    int e = blockIdx.x * blockDim.x + threadIdx.x;
    if (e < nEdges) atomicAdd(&rowCnt[rows[e]], 1);
}

// Single-workgroup chunked Hillis-Steele exclusive scan over nNodes counts.
// 100k elements / 1024 threads -> ~98 chunks; trivial runtime.
__global__ __launch_bounds__(1024) void scan_counts(
    const int* __restrict__ rowCnt, int* __restrict__ rowOff,
    int* __restrict__ cursor, int n)
{
    __shared__ int smem[1024];
    const int tid = threadIdx.x;
    int running = 0;                         // identical in every thread

    for (int base = 0; base < n; base += 1024) {
        const int i = base + tid;
        const int v = (i < n) ? rowCnt[i] : 0;
        smem[tid] = v;
        __syncthreads();
        #pragma unroll
        for (int d = 1; d < 1024; d <<= 1) {
            int t = (tid >= d) ? smem[tid - d] : 0;
            __syncthreads();
            smem[tid] += t;
            __syncthreads();
        }
        const int excl = smem[tid] - v;      // exclusive within chunk
        if (i < n) {
            rowOff[i] = running + excl;
            cursor[i] = running + excl;
        }
        running += smem[1023];               // chunk total (all threads agree)
        __syncthreads();                     // protect smem for next chunk
    }
    if (tid == 0) rowOff[n] = running;       // == nEdges
}

__global__ __launch_bounds__(256) void fill_perm(
    const int* __restrict__ rows, int* __restrict__ cursor,
    int* __restrict__ perm, int nEdges)
{
    int e = blockIdx.x * blockDim.x + threadIdx.x;
    if (e < nEdges) {
        int p = atomicAdd(&cursor[rows[e]], 1);
        perm[p] = e;
    }
}

// ---------------------------------------------------------------------------
// Gather SpMM: one wave per output row. Lane l accumulates a float4 chunk in
// registers over the row's edges; support rows hit L2 (51 MB << 192 MB L2).
// Bias folded into accumulator init; single coalesced b128 store per lane.
// ---------------------------------------------------------------------------
__global__ __launch_bounds__(256) void spmm_gather(
    const int* __restrict__ rowOff, const int* __restrict__ perm,
    const int* __restrict__ cols, const float* __restrict__ vals,
    const float* __restrict__ support, const float* __restrict__ bias,
    float* __restrict__ out, int nNodes)
{
    const int lane   = threadIdx.x & 31;
    const int wave   = (blockIdx.x * blockDim.x + threadIdx.x) >> 5;
    const int nWaves = (gridDim.x * blockDim.x) >> 5;

    const float4 b4 = ((const float4*)bias)[lane];

    for (int r = wave; r < nNodes; r += nWaves) {
        const int beg = rowOff[r];
        const int end = rowOff[r + 1];

        float4 acc = b4;
        for (int i = beg; i < end; ++i) {
            const int   e = perm[i];
            const int   c = cols[e];
            const float v = vals[e];
            const float4 s = ((const float4*)(support + (size_t)c * DF))[lane];
            acc.x += v * s.x;
            acc.y += v * s.y;
            acc.z += v * s.z;
            acc.w += v * s.w;
        }
        ((float4*)(out + (size_t)r * DF))[lane] = acc;
    }
}

// ---------------------------------------------------------------------------
// Fallback (small workspace): bias-init + edge-parallel atomic scatter.
// ---------------------------------------------------------------------------
__global__ __launch_bounds__(256) void init_out(
    float* __restrict__ out, const float* __restrict__ bias, int n)
{
    int i = blockIdx.x * blockDim.x + threadIdx.x;
    if (i < n) out[i] = bias[i & (DF - 1)];
}

__global__ __launch_bounds__(256) void spmm_atomic(
    const int* __restrict__ rows, const int* __restrict__ cols,
    const float* __restrict__ vals, const float* __restrict__ support,
    float* __restrict__ out, int nEdges)
{
    const int lane   = threadIdx.x & 31;
    const int wave   = (blockIdx.x * blockDim.x + threadIdx.x) >> 5;
    const int nWaves = (gridDim.x * blockDim.x) >> 5;

    for (int e = wave; e < nEdges; e += nWaves) {
        const int   r = rows[e];
        const int   c = cols[e];
        const float v = vals[e];
        const float4 s = ((const float4*)(support + (size_t)c * DF))[lane];
        float* dst = out + (size_t)r * DF + (lane << 2);
        atomicAdd(dst + 0, v * s.x);
        atomicAdd(dst + 1, v * s.y);
        atomicAdd(dst + 2, v * s.z);
        atomicAdd(dst + 3, v * s.w);
    }
}

// ---------------------------------------------------------------------------
extern "C" void kernel_launch(void* const* d_in, const int* in_sizes, int n_in,
                              void* d_out, int out_size, void* d_ws, size_t ws_size,
                              hipStream_t stream)
{
    const float* x        = (const float*)d_in[0];
    const float* W        = (const float*)d_in[1];
    const float* lin_bias = (const float*)d_in[2];
    const float* bias     = (const float*)d_in[3];
    const int*   rows     = (const int*)d_in[4];
    const int*   cols     = (const int*)d_in[5];
    const float* vals     = (const float*)d_in[6];

    float* out = (float*)d_out;

    const int nNodes = in_sizes[0] / DF;
    const int nEdges = in_sizes[4];
    const int mTiles = (nNodes + 15) / 16;

    // Workspace layout
    float* support = (float*)d_ws;                              // nNodes*128 f32
    size_t supBytes = (size_t)nNodes * DF * sizeof(float);
    int* rowCnt = (int*)((char*)d_ws + supBytes);               // nNodes+1
    int* rowOff = rowCnt + (nNodes + 1);                        // nNodes+1
    int* cursor = rowOff + (nNodes + 1);                        // nNodes
    int* perm   = cursor + nNodes;                              // nEdges
    size_t csrBytes = supBytes +
        (size_t)(3 * (nNodes + 1) + nEdges) * sizeof(int);

    // 1) support = x @ W + lin_bias  (WMMA)
    const int blocksG = (mTiles + 7) / 8;
    gemm_bias_wmma<<<blocksG, 256, 0, stream>>>(x, W, lin_bias, support,
                                                nNodes, mTiles);

    const int eBlocks = (nEdges + 255) / 256;

    if (ws_size >= csrBytes) {
        // 2) CSR build: histogram -> scan -> permutation
        zero_ints<<<(nNodes + 256) / 256, 256, 0, stream>>>(rowCnt, nNodes + 1);
        hist_rows<<<eBlocks, 256, 0, stream>>>(rows, rowCnt, nEdges);
        scan_counts<<<1, 1024, 0, stream>>>(rowCnt, rowOff, cursor, nNodes);
        fill_perm<<<eBlocks, 256, 0, stream>>>(rows, cursor, perm, nEdges);

        // 3) gather: one wave per row, no atomics in hot phase
        const int rBlocks = (nNodes + 7) / 8;       // 8 waves per block
        spmm_gather<<<rBlocks, 256, 0, stream>>>(rowOff, perm, cols, vals,
                                                 support, bias, out, nNodes);
    } else {
        // Fallback: bias init + atomic scatter
        init_out<<<(out_size + 255) / 256, 256, 0, stream>>>(out, bias, out_size);
        spmm_atomic<<<2048, 256, 0, stream>>>(rows, cols, vals, support, out,
                                              nEdges);
    }
}